// GCNEncoder_51230369906741
// MI455X (gfx1250) — compile-verified
//
#include <hip/hip_runtime.h>
#include <math.h>

// GCN encoder for MI455X (gfx1250, wave32).
//   layer1: H1 = X @ W1^T            (WMMA f32 16x16x4, K=128)
//   agg1  : A1 = b1 + scatter_add(norm * H1[src] -> dst)   (global_atomic_add_f32)
//   BN+ReLU fused into GEMM2 A-load
//   layer2: H2 = relu(bn(A1)) @ W2^T (WMMA f32 16x16x4)
//   agg2  : out = b2 + scatter_add(norm * H2[src] -> dst)

#define FEAT1 128
#define FEAT2 64
#define BN_EPS 1e-5f

typedef float v2f __attribute__((ext_vector_type(2)));
typedef float v8f __attribute__((ext_vector_type(8)));

// Fire-and-forget HW float atomic at device scope (L2 atomic units).
__device__ __forceinline__ void gatomic_add_f32(float* p, float v) {
    asm volatile("global_atomic_add_f32 %0, %1, off scope:SCOPE_DEV"
                 :: "v"(p), "v"(v) : "memory");
}

// ---------------- init / small elementwise kernels ----------------

__global__ void fill_f32(float* __restrict__ p, float v, size_t n) {
    size_t i = (size_t)blockIdx.x * blockDim.x + threadIdx.x;
    if (i < n) p[i] = v;
}

// p[i] = bias[i & mask]   (F is a power of two)
__global__ void bcast_bias(float* __restrict__ p, const float* __restrict__ b,
                           int mask, size_t n) {
    size_t i = (size_t)blockIdx.x * blockDim.x + threadIdx.x;
    if (i < n) p[i] = b[i & (size_t)mask];
}

// deg[dst_e] += 1  over real edges (self-loop "1" pre-seeded by fill_f32)
__global__ void deg_accum(const long long* __restrict__ ei, float* __restrict__ deg, int E) {
    int e = blockIdx.x * blockDim.x + threadIdx.x;
    if (e < E) {
        int d = (int)ei[(size_t)E + e];
        gatomic_add_f32(&deg[d], 1.0f);
    }
}

__global__ void deg_to_dis(float* __restrict__ deg, int N) {
    int i = blockIdx.x * blockDim.x + threadIdx.x;
    if (i < N) deg[i] = rsqrtf(deg[i]);   // deg >= 1 always (self-loops)
}

// ---------------- WMMA GEMM:  H[M x FOUT] = A[M x 128] @ W[FOUT x 128]^T ----------------
// One wave per 16x16 output tile; blockDim = (32, FOUT/16); K = 128 in 32 steps of 4.
// FUSE_BN: A-element -> relu(a * ss[k] + ss[128+k]) on the fly.

template <int FOUT, bool FUSE_BN>
__launch_bounds__(256)
__global__ void gemm_wmma_f32(const float* __restrict__ A,
                              const float* __restrict__ W,
                              float* __restrict__ H,
                              const float* __restrict__ ss,
                              int M) {
    const int lane = threadIdx.x;        // 0..31
    const int half = lane >> 4;          // 0 or 1
    const int r    = lane & 15;
    const int m0   = blockIdx.x * 16;
    const int n0   = threadIdx.y * 16;

    int mr = m0 + r;
    if (mr >= M) mr = M - 1;             // clamp (harmless duplicate read)
    const float* arow = A + (size_t)mr * FEAT1;
    const float* wrow = W + (size_t)(n0 + r) * FEAT1;

    v8f acc = {0.f, 0.f, 0.f, 0.f, 0.f, 0.f, 0.f, 0.f};

#pragma unroll
    for (int k0 = 0; k0 < FEAT1; k0 += 4) {
        const int ka = k0 + half * 2;
        // A frag: lane r holds row m0+r, K = ka, ka+1  (contiguous 8B load)
        v2f a = *(const v2f*)(arow + ka);
        // B frag: B[k][n] = W[n0+n][k]; lane r supplies column n0+r, K = ka, ka+1
        v2f b = *(const v2f*)(wrow + ka);
        if (FUSE_BN) {
            a.x = fmaxf(0.0f, fmaf(a.x, ss[ka],     ss[FEAT1 + ka]));
            a.y = fmaxf(0.0f, fmaf(a.y, ss[ka + 1], ss[FEAT1 + ka + 1]));
        }
        acc = __builtin_amdgcn_wmma_f32_16x16x4_f32(
                  false, a, false, b, (short)0, acc, false, false);
    }

    // C/D layout: VGPR v -> M = m0 + half*8 + v, N = n0 + (lane&15)
#pragma unroll
    for (int v = 0; v < 8; ++v) {
        int row = m0 + half * 8 + v;
        if (row < M) H[(size_t)row * FOUT + n0 + r] = acc[v];
    }
}

// ---------------- edge aggregation: out[dst] += norm * H[src] ----------------
// One wave per (edge or self-loop). F=128: float4/lane; F=64: float2/lane.

template <int F>
__launch_bounds__(256)
__global__ void aggregate(const float* __restrict__ H,
                          const long long* __restrict__ ei,
                          const float* __restrict__ dis,
                          float* __restrict__ out,
                          int E, int N) {
    const int e = blockIdx.x * blockDim.y + threadIdx.y;   // wave-uniform
    if (e >= E + N) return;
    const int lane = threadIdx.x;

    int s, d;
    float nrm;
    if (e < E) {
        s = (int)ei[e];
        d = (int)ei[(size_t)E + e];
        nrm = dis[s] * dis[d];
    } else {                 // appended self-loop
        s = d = e - E;
        float t = dis[s];
        nrm = t * t;
    }

    if (F == 128) {
        const float4 h = *(const float4*)(H + (size_t)s * 128 + lane * 4);
        float* o = out + (size_t)d * 128 + lane * 4;
        gatomic_add_f32(o + 0, h.x * nrm);
        gatomic_add_f32(o + 1, h.y * nrm);
        gatomic_add_f32(o + 2, h.z * nrm);
        gatomic_add_f32(o + 3, h.w * nrm);
    } else {
        const float2 h = *(const float2*)(H + (size_t)s * 64 + lane * 2);
        float* o = out + (size_t)d * 64 + lane * 2;
        gatomic_add_f32(o + 0, h.x * nrm);
        gatomic_add_f32(o + 1, h.y * nrm);
    }
}

// ---------------- BatchNorm statistics ----------------
// blockDim = 128 (thread f owns feature f); rows strided across blocks.

__global__ void bn_stats(const float* __restrict__ A, float* __restrict__ stats, int N) {
    const int f = threadIdx.x;           // 0..127
    float s = 0.0f, s2 = 0.0f;
    for (int n = blockIdx.x; n < N; n += gridDim.x) {
        float v = A[(size_t)n * FEAT1 + f];
        s  += v;
        s2 += v * v;
    }
    gatomic_add_f32(&stats[f], s);
    gatomic_add_f32(&stats[FEAT1 + f], s2);
}

__global__ void bn_finalize(const float* __restrict__ stats,
                            const float* __restrict__ gamma,
                            const float* __restrict__ beta,
                            float* __restrict__ ss, int N) {
    const int f = threadIdx.x;           // 0..127
    const float inv = 1.0f / (float)N;
    float mean = stats[f] * inv;
    float var  = stats[FEAT1 + f] * inv - mean * mean;
    float sc   = gamma[f] * rsqrtf(var + BN_EPS);
    ss[f]         = sc;
    ss[FEAT1 + f] = beta[f] - mean * sc;
}

// ---------------- launch ----------------

extern "C" void kernel_launch(void* const* d_in, const int* in_sizes, int n_in,
                              void* d_out, int out_size, void* d_ws, size_t ws_size,
                              hipStream_t stream) {
    const float*     x     = (const float*)d_in[0];
    const long long* ei    = (const long long*)d_in[1];   // int64 [2, E]
    const float*     W1    = (const float*)d_in[2];       // [128,128]
    const float*     b1    = (const float*)d_in[3];
    const float*     gamma = (const float*)d_in[4];
    const float*     beta  = (const float*)d_in[5];
    const float*     W2    = (const float*)d_in[6];       // [64,128]
    const float*     b2    = (const float*)d_in[7];
    float*           out   = (float*)d_out;               // [N, 64]

    const int N = in_sizes[0] / FEAT1;
    const int E = in_sizes[1] / 2;

    // Workspace layout (floats); all segment sizes are 16B-multiples.
    float* dis   = (float*)d_ws;                 // N        (deg -> 1/sqrt(deg))
    float* H1    = dis + N;                      // N*128
    float* A1    = H1 + (size_t)N * FEAT1;       // N*128    (b1 + aggregated, BN input)
    float* H2    = A1 + (size_t)N * FEAT1;       // N*64
    float* stats = H2 + (size_t)N * FEAT2;       // 256      (sum, sumsq)
    float* ss    = stats + 256;                  // 256      (scale, shift)

    const int mtiles = (N + 15) / 16;
    const int aggblk = (E + N + 7) / 8;          // 8 waves / block

    // degrees: seed 1.0 (self-loop), accumulate edges, rsqrt in place
    fill_f32<<<(N + 255) / 256, 256, 0, stream>>>(dis, 1.0f, (size_t)N);
    deg_accum<<<(E + 255) / 256, 256, 0, stream>>>(ei, dis, E);
    deg_to_dis<<<(N + 255) / 256, 256, 0, stream>>>(dis, N);

    // layer 1 GEMM (WMMA f32)
    gemm_wmma_f32<FEAT1, false><<<mtiles, dim3(32, 8), 0, stream>>>(x, W1, H1, nullptr, N);

    // A1 = b1 broadcast, then scatter-add messages
    bcast_bias<<<(int)(((size_t)N * FEAT1 + 255) / 256), 256, 0, stream>>>(
        A1, b1, FEAT1 - 1, (size_t)N * FEAT1);
    aggregate<FEAT1><<<aggblk, dim3(32, 8), 0, stream>>>(H1, ei, dis, A1, E, N);

    // BatchNorm statistics -> scale/shift
    fill_f32<<<1, 256, 0, stream>>>(stats, 0.0f, (size_t)256);
    bn_stats<<<1024, FEAT1, 0, stream>>>(A1, stats, N);
    bn_finalize<<<1, FEAT1, 0, stream>>>(stats, gamma, beta, ss, N);

    // layer 2 GEMM with fused BN + ReLU on A
    gemm_wmma_f32<FEAT2, true><<<mtiles, dim3(32, 4), 0, stream>>>(A1, W2, H2, ss, N);

    // out = b2 broadcast, then scatter-add messages
    bcast_bias<<<(int)(((size_t)N * FEAT2 + 255) / 256), 256, 0, stream>>>(
        out, b2, FEAT2 - 1, (size_t)N * FEAT2);
    aggregate<FEAT2><<<aggblk, dim3(32, 8), 0, stream>>>(H2, ei, dis, out, E, N);
}